// KnowledgeSheaf_27522150433500
// MI455X (gfx1250) — compile-verified
//
#include <hip/hip_runtime.h>
#include <math.h>

// KnowledgeSheaf on MI455X (gfx1250, wave32, WMMA).
//
// Reformulation: comparison[v] = (Σ_t c_v[t] R[τ_v,t]) x_v with
//   c_v[t] = outdeg_by_dst_type(v,t) - indeg_by_src_type(v,t)
// => edge pass = 2 int atomics/edge; node pass = 16x16x256 f32 GEMM per
// 16-node same-type group via V_WMMA_F32_16X16X4_F32 (K=4, 64 WMMAs/group).
//
// v2: added coalesced transpose+scale pass (xt[v][16] = reps[:,v]*inv_deg[v])
// so the WMMA pass loads node vectors as b128 from a single line instead of a
// 16-way strided gather; R-map LDS staging upgraded to float4 / ds_store_b128.
//
// Workspace need: ~7 MB.

#define STALK 16
#define NTYPES 16

typedef __attribute__((ext_vector_type(2))) float v2f;
typedef __attribute__((ext_vector_type(8))) float v8f;

#define ASTRIDE 260   // 256 + 4 pad: avoids 16-way LDS bank conflict on A rows
                      // (ASTRIDE*4 bytes = 1040, 16B-aligned => float4 stores OK)

// ---------------- init ----------------
__global__ void ks_init(unsigned* udeg, int* cnt, int* chunkHist, int* sortedIds,
                        int N, int nHist, int padCap) {
    int i = blockIdx.x * blockDim.x + threadIdx.x;
    int total = N * NTYPES;            // largest region
    if (i < total)  cnt[i] = 0;
    if (i < N)      udeg[i] = 0u;
    if (i < nHist)  chunkHist[i] = 0;
    if (i < padCap) sortedIds[i] = -1;
}

// ---------------- edge pass: degree + signed type counts ----------------
__global__ void ks_edges(const int* __restrict__ ei, const int* __restrict__ types,
                         unsigned* udeg, int* cnt, int E) {
    int e = blockIdx.x * blockDim.x + threadIdx.x;
    if (e >= E) return;
    int s = ei[e];
    int d = ei[E + e];
    atomicAdd(&udeg[d], 1u);
    int ts = types[s];
    int td = types[d];
    atomicAdd(&cnt[s * NTYPES + td],  1);
    atomicAdd(&cnt[d * NTYPES + ts], -1);
}

// ---------------- inv_deg = deg>0 ? rsqrt(deg) : 0 (in place) ----------------
__global__ void ks_invdeg(unsigned* udeg, float* inv, int N) {
    int v = blockIdx.x * blockDim.x + threadIdx.x;
    if (v >= N) return;
    unsigned u = udeg[v];
    inv[v] = (u > 0u) ? rsqrtf((float)u) : 0.0f;
}

// ---------------- transpose + scale: xt[v][j] = reps[j][v] * inv[v] ----------------
// Reads coalesced per j (consecutive v across lanes); each thread writes its
// node's 16 floats contiguously (64B) -> fully coalesced both directions.
__global__ void ks_transpose(const float* __restrict__ reps,
                             const float* __restrict__ inv,
                             float* __restrict__ xt, int N) {
    int v = blockIdx.x * blockDim.x + threadIdx.x;
    if (v >= N) return;
    float s = inv[v];
    float buf[16];
    #pragma unroll
    for (int j = 0; j < 16; ++j) buf[j] = reps[j * N + v] * s;
    float4* dst = (float4*)&xt[(size_t)v * 16];
    #pragma unroll
    for (int q = 0; q < 4; ++q)
        dst[q] = make_float4(buf[4*q], buf[4*q+1], buf[4*q+2], buf[4*q+3]);
}

// ---------------- deterministic type binning ----------------
__global__ void ks_chunk_hist(const int* __restrict__ types, int* chunkHist, int N) {
    int i = blockIdx.x * blockDim.x + threadIdx.x;
    if (i >= N) return;
    atomicAdd(&chunkHist[(i >> 8) * NTYPES + types[i]], 1);
}

__global__ void ks_scan(const int* __restrict__ chunkHist, int* chunkBase,
                        int* typeOff, int nChunks) {
    __shared__ int typeTotal[NTYPES];
    int t = threadIdx.x;
    if (t < NTYPES) {
        int run = 0;
        for (int c = 0; c < nChunks; ++c) {
            chunkBase[c * NTYPES + t] = run;
            run += chunkHist[c * NTYPES + t];
        }
        typeTotal[t] = run;
    }
    __syncthreads();
    if (t == 0) {
        int run = 0;
        for (int k = 0; k < NTYPES; ++k) {
            typeOff[k] = run;
            run += (typeTotal[k] + 15) & ~15;   // 16-align each type segment
        }
        typeOff[NTYPES] = run;
    }
}

__global__ void ks_scatter(const int* __restrict__ types,
                           const int* __restrict__ chunkBase,
                           const int* __restrict__ typeOff,
                           int* sortedIds, int N) {
    __shared__ int lt[256];
    int chunk = blockIdx.x;
    int node = chunk * 256 + threadIdx.x;
    int myt = 0xFF;
    if (node < N) myt = types[node];
    lt[threadIdx.x] = myt;
    __syncthreads();
    if (node >= N) return;
    int rank = 0;
    for (int j = 0; j < (int)threadIdx.x; ++j)
        rank += (lt[j] == myt) ? 1 : 0;
    int pos = typeOff[myt] + chunkBase[chunk * NTYPES + myt] + rank;
    sortedIds[pos] = node;
}

// ---------------- WMMA node pass: Y(16x16) = R~[tau](16x256) @ W(256x16) ----------------
__global__ __launch_bounds__(32)
void ks_wmma(const float* __restrict__ xt,        // [N, 16] scaled reps
             const float* __restrict__ maps,      // [16,16,16,16]
             const int*   __restrict__ types,
             const int*   __restrict__ cnt,       // [N,16] signed
             const int*   __restrict__ sortedIds,
             float* __restrict__ partials) {
    __shared__ float Alds[STALK * ASTRIDE];       // A[i][q], q = t*16+j
    __shared__ float Wlds[256 * 16];              // W[q][n]
    __shared__ int   ids_s[16];

    const int lane = threadIdx.x;
    const int base = blockIdx.x * 16;

    if (lane < 16) ids_s[lane] = sortedIds[base + lane];
    __syncthreads();

    if (ids_s[0] < 0) {                           // fully-padded group
        if (lane == 0) partials[blockIdx.x] = 0.0f;
        return;
    }
    const int tau = types[ids_s[0]];
    const float* Rt = maps + (size_t)tau * 4096;  // R[tau][t][i][j]

    // Stage A = R~[tau]: A[i][t*16+j] = Rt[t*256 + i*16 + j]
    // float4 global loads (coalesced b128) + ds_store_b128 (j stays in-row).
    for (int m4 = lane * 4; m4 < 4096; m4 += 128) {
        int t = m4 >> 8, i = (m4 >> 4) & 15, j = m4 & 15;
        float4 val = *(const float4*)&Rt[m4];
        *(float4*)&Alds[i * ASTRIDE + t * 16 + j] = val;
    }

    // Build W: W[t*16+j][n] = c_n[t] * x_n[j]; lane = n + 16*half, half owns 8 t's
    const int n     = lane & 15;
    const int id    = ids_s[n];
    const int tbase = (lane >> 4) * 8;
    if (id >= 0) {
        float xv[16];
        const float4* xp = (const float4*)&xt[(size_t)id * 16];
        #pragma unroll
        for (int q = 0; q < 4; ++q) {
            float4 x4 = xp[q];
            xv[4*q+0] = x4.x; xv[4*q+1] = x4.y; xv[4*q+2] = x4.z; xv[4*q+3] = x4.w;
        }
        const int4* cp = (const int4*)&cnt[(size_t)id * NTYPES + tbase];
        int cc[8];
        #pragma unroll
        for (int q = 0; q < 2; ++q) {
            int4 c4 = cp[q];
            cc[4*q+0] = c4.x; cc[4*q+1] = c4.y; cc[4*q+2] = c4.z; cc[4*q+3] = c4.w;
        }
        #pragma unroll
        for (int tt = 0; tt < 8; ++tt) {
            int t = tbase + tt;
            float c = (float)cc[tt];
            #pragma unroll
            for (int j = 0; j < 16; ++j)
                Wlds[(t * 16 + j) * 16 + n] = c * xv[j];
        }
    } else {
        #pragma unroll
        for (int tt = 0; tt < 8; ++tt) {
            int t = tbase + tt;
            #pragma unroll
            for (int j = 0; j < 16; ++j)
                Wlds[(t * 16 + j) * 16 + n] = 0.0f;
        }
    }
    __syncthreads();

    // 16x16x256 GEMM as 64 accumulating V_WMMA_F32_16X16X4_F32.
    // A 16x4 layout: lanes 0-15 rows M, vgpr{0,1}=K{0,1}; lanes 16-31 K{2,3}.
    // B 4x16 layout: lanes 0-15 cols N, vgpr{0,1}=K{0,1}; lanes 16-31 K{2,3}.
    const int row  = lane & 15;
    const int koff = (lane >> 4) * 2;
    const float* Arow = Alds + row * ASTRIDE;
    v8f acc = {0.f, 0.f, 0.f, 0.f, 0.f, 0.f, 0.f, 0.f};
    #pragma unroll 8
    for (int k = 0; k < 64; ++k) {
        const int q = 4 * k;
        v2f a, b;
        a.x = Arow[q + koff];
        a.y = Arow[q + koff + 1];
        b.x = Wlds[(q + koff)     * 16 + row];
        b.y = Wlds[(q + koff + 1) * 16 + row];
        acc = __builtin_amdgcn_wmma_f32_16x16x4_f32(
                  /*neg_a=*/false, a, /*neg_b=*/false, b,
                  /*c_mod=*/(short)0, acc, /*reuse_a=*/false, /*reuse_b=*/false);
    }

    // D layout: lane<16 vgpr r -> Y[r][col=row]; lane>=16 -> Y[r+8][col=row].
    float local = 0.0f;
    if (id >= 0) {
        #pragma unroll
        for (int r = 0; r < 8; ++r) local += acc[r] * acc[r];
    }
    #pragma unroll
    for (int off = 16; off > 0; off >>= 1)
        local += __shfl_down(local, off, 32);
    if (lane == 0) partials[blockIdx.x] = local;
}

// ---------------- deterministic final reduction ----------------
__global__ void ks_reduce(const float* __restrict__ partials, int n, float* out) {
    __shared__ float s[256];
    float acc = 0.0f;
    for (int i = threadIdx.x; i < n; i += 256) acc += partials[i];
    s[threadIdx.x] = acc;
    __syncthreads();
    for (int st = 128; st > 0; st >>= 1) {
        if ((int)threadIdx.x < st) s[threadIdx.x] += s[threadIdx.x + st];
        __syncthreads();
    }
    if (threadIdx.x == 0) out[0] = s[0];
}

extern "C" void kernel_launch(void* const* d_in, const int* in_sizes, int n_in,
                              void* d_out, int out_size, void* d_ws, size_t ws_size,
                              hipStream_t stream) {
    const float* reps  = (const float*)d_in[0];   // [16, N]
    const float* maps  = (const float*)d_in[1];   // [16,16,16,16]
    const int*   ei    = (const int*)  d_in[2];   // [2, E]
    const int*   types = (const int*)  d_in[3];   // [N]

    const int N = in_sizes[3];
    const int E = in_sizes[2] / 2;

    const int nChunks = (N + 255) / 256;
    const int nHist   = nChunks * NTYPES;
    const int padCap  = (((N + NTYPES * 16) + 15) / 16) * 16;  // all segments 16-aligned
    const int nGroups = padCap / 16;

    // ---- carve workspace (floats/int32 words) ----
    float* wsf = (float*)d_ws;
    size_t o = 0;
    float*    inv       = wsf + o;            o += (size_t)N;           // also udeg during edge pass
    int*      cnt       = (int*)(wsf + o);    o += (size_t)N * NTYPES;
    float*    xt        = wsf + o;            o += (size_t)N * STALK;   // [N,16] scaled reps
    int*      chunkHist = (int*)(wsf + o);    o += (size_t)nHist;
    int*      chunkBase = (int*)(wsf + o);    o += (size_t)nHist;
    int*      typeOff   = (int*)(wsf + o);    o += 32;
    int*      sortedIds = (int*)(wsf + o);    o += (size_t)padCap;
    float*    partials  = wsf + o;            o += (size_t)nGroups;
    unsigned* udeg      = (unsigned*)inv;
    (void)ws_size; (void)n_in; (void)out_size;

    const int TB = 256;
    // 1) init
    {
        int total = N * NTYPES;   // covers all smaller regions too
        ks_init<<<(total + TB - 1) / TB, TB, 0, stream>>>(udeg, cnt, chunkHist,
                                                          sortedIds, N, nHist, padCap);
    }
    // 2) edge pass
    ks_edges<<<(E + TB - 1) / TB, TB, 0, stream>>>(ei, types, udeg, cnt, E);
    // 3) inv_deg
    ks_invdeg<<<(N + TB - 1) / TB, TB, 0, stream>>>(udeg, inv, N);
    // 4) transpose + scale node reps into [N,16]
    ks_transpose<<<(N + TB - 1) / TB, TB, 0, stream>>>(reps, inv, xt, N);
    // 5) deterministic binning by type
    ks_chunk_hist<<<(N + TB - 1) / TB, TB, 0, stream>>>(types, chunkHist, N);
    ks_scan<<<1, 32, 0, stream>>>(chunkHist, chunkBase, typeOff, nChunks);
    ks_scatter<<<nChunks, 256, 0, stream>>>(types, chunkBase, typeOff, sortedIds, N);
    // 6) WMMA node pass (one wave per 16-node type-uniform group)
    ks_wmma<<<nGroups, 32, 0, stream>>>(xt, maps, types, cnt, sortedIds, partials);
    // 7) deterministic scalar reduction
    ks_reduce<<<1, 256, 0, stream>>>(partials, nGroups, (float*)d_out);
}